// SAE_9835475107848
// MI455X (gfx1250) — compile-verified
//
#include <hip/hip_runtime.h>
#include <hip/hip_bf16.h>

#define DMODEL 2048
#define NFEAT  32768
#define BATCH  2048
#define TOPK   64
#define AUXK   32
#define DEAD_AFTER 1000
#define AUXK_AFTER 256

typedef __attribute__((ext_vector_type(2))) float v2f;
typedef __attribute__((ext_vector_type(8))) float v8f;

// CDNA5 async memory->LDS copy (ISA §15.18.3 op 98), tracked by ASYNCcnt.
__device__ __forceinline__ void async_load_lds_b128(unsigned lds_off, const void* gaddr) {
    asm volatile("global_load_async_to_lds_b128 %0, %1, off"
                 :: "v"(lds_off), "v"(gaddr) : "memory");
}
__device__ __forceinline__ void wait_asynccnt0() {
    asm volatile("s_wait_asynccnt 0x0" ::: "memory");
}
__device__ __forceinline__ unsigned lds_offset(const void* p) {
    // Generic (flat) LDS pointers keep the workgroup-relative LDS byte offset
    // in addr[31:0] (ISA §10.2 aperture mapping).
    return (unsigned)(unsigned long long)p;
}

// ---------------------------------------------------------------------------
// Kernel 0: c[k] = b_pre[k] + b_post[k]
// ---------------------------------------------------------------------------
__global__ __launch_bounds__(256) void sae_prep(
    const float* __restrict__ b_pre, const float* __restrict__ b_post,
    float* __restrict__ c)
{
    int i = blockIdx.x * 256 + threadIdx.x;
    if (i < DMODEL) c[i] = b_pre[i] + b_post[i];
}

// ---------------------------------------------------------------------------
// Kernel 1: encodings = (x*scale/avg_norm - c) @ W_enc
// f32 WMMA 16x16x4, block tile 256x128, BK=16, double-buffered LDS.
// B tile staged with GLOBAL_LOAD_ASYNC_TO_LDS_B128; A tile via VGPRs
// (needs scale/bias transform + k-major transpose).
// ---------------------------------------------------------------------------
__global__ __launch_bounds__(256) void sae_enc_gemm(
    const float* __restrict__ x, const float* __restrict__ W_enc,
    const float* __restrict__ c, const float* __restrict__ avg_norm,
    float* __restrict__ enc)
{
    __shared__ float As[2][16 * 256];   // k-major: As[k][m], 32KB
    __shared__ float Bs[2][16 * 128];   // row-major: Bs[k][n], 16KB

    const int tid  = threadIdx.x;
    const int lane = tid & 31;
    const int wid  = tid >> 5;      // 8 waves
    const int wvm  = wid >> 1;      // 0..3 -> wave M row (64 rows each)
    const int wvn  = wid & 1;       // 0..1 -> wave N col (64 cols each)
    const int half = lane >> 4;     // 0..1
    const int lm   = lane & 15;     // 0..15
    const int bm   = blockIdx.y * 256;
    const int bn   = blockIdx.x * 128;
    const float s  = sqrtf((float)DMODEL) / avg_norm[0];

    v8f acc[4][4];
#pragma unroll
    for (int i = 0; i < 4; ++i)
#pragma unroll
        for (int j = 0; j < 4; ++j)
            acc[i][j] = (v8f){0.f, 0.f, 0.f, 0.f, 0.f, 0.f, 0.f, 0.f};

    float4 ra[4];

    // A tile: 256(m) x 16(k); per thread 4 float4 chunks along k.
    auto loadA = [&](int k0) {
#pragma unroll
        for (int u = 0; u < 4; ++u) {
            int f  = tid + u * 256;           // 0..1023
            int ka = (f & 3) * 4;
            int m  = f >> 2;                  // 0..255
            float4 v  = *(const float4*)(x + (size_t)(bm + m) * DMODEL + k0 + ka);
            float4 cv = *(const float4*)(c + k0 + ka);
            v.x = v.x * s - cv.x;
            v.y = v.y * s - cv.y;
            v.z = v.z * s - cv.z;
            v.w = v.w * s - cv.w;
            ra[u] = v;
        }
    };
    auto storeA = [&](int buf) {
#pragma unroll
        for (int u = 0; u < 4; ++u) {
            int f  = tid + u * 256;
            int ka = (f & 3) * 4;
            int m  = f >> 2;
            As[buf][(ka + 0) * 256 + m] = ra[u].x;
            As[buf][(ka + 1) * 256 + m] = ra[u].y;
            As[buf][(ka + 2) * 256 + m] = ra[u].z;
            As[buf][(ka + 3) * 256 + m] = ra[u].w;
        }
    };
    // B tile: 16(k) x 128(n); pure copy -> async direct to LDS.
    auto asyncB = [&](int k0, int buf) {
#pragma unroll
        for (int u = 0; u < 2; ++u) {
            int f  = tid + u * 256;           // 0..511
            int na = (f & 31) * 4;
            int kb = f >> 5;                  // 0..15
            async_load_lds_b128(lds_offset(&Bs[buf][kb * 128 + na]),
                                W_enc + (size_t)(k0 + kb) * NFEAT + bn + na);
        }
    };

    auto compute = [&](int buf) {
#pragma unroll
        for (int kk = 0; kk < 16; kk += 4) {
            // A frag (16x4 f32): lane(half,lm) -> M = m0+lm, frag[v]: K = kk + v + 2*half
            v2f a[4], b[4];
#pragma unroll
            for (int t = 0; t < 4; ++t) {
                int m = wvm * 64 + t * 16 + lm;
                a[t].x = As[buf][(kk + 2 * half + 0) * 256 + m];
                a[t].y = As[buf][(kk + 2 * half + 1) * 256 + m];
            }
#pragma unroll
            for (int t = 0; t < 4; ++t) {
                int n = wvn * 64 + t * 16 + lm;
                b[t].x = Bs[buf][(kk + 2 * half + 0) * 128 + n];
                b[t].y = Bs[buf][(kk + 2 * half + 1) * 128 + n];
            }
#pragma unroll
            for (int i = 0; i < 4; ++i)
#pragma unroll
                for (int j = 0; j < 4; ++j)
                    acc[i][j] = __builtin_amdgcn_wmma_f32_16x16x4_f32(
                        false, a[i], false, b[j], (short)0, acc[i][j], false, false);
        }
    };

    asyncB(0, 0);
    loadA(0);
    storeA(0);
    wait_asynccnt0();
    __syncthreads();

    int buf = 0;
    for (int t = 0; t < DMODEL / 16; ++t) {
        const bool hasNext = (t + 1) < (DMODEL / 16);
        if (hasNext) {
            asyncB((t + 1) * 16, buf ^ 1);    // HW copies while we compute
            loadA((t + 1) * 16);
        }
        compute(buf);
        if (hasNext) storeA(buf ^ 1);
        wait_asynccnt0();                     // our async B writes for buf^1 done
        __syncthreads();
        buf ^= 1;
    }

    // C/D layout: VGPR r, lanes 0-15 -> M=r, N=lane; lanes 16-31 -> M=8+r, N=lane-16
#pragma unroll
    for (int i = 0; i < 4; ++i)
#pragma unroll
        for (int j = 0; j < 4; ++j) {
            int n = bn + wvn * 64 + j * 16 + lm;
#pragma unroll
            for (int r = 0; r < 8; ++r) {
                int m = bm + wvm * 64 + i * 16 + r + 8 * half;
                enc[(size_t)m * NFEAT + n] = acc[i][j][r];
            }
        }
}

// ---------------------------------------------------------------------------
// Kernel 2: exact per-row top-64 via 4-pass radix select + bitonic sort.
// ---------------------------------------------------------------------------
__device__ __forceinline__ unsigned fkey(float f) {
    unsigned u = __float_as_uint(f);
    return u ^ ((unsigned)((int)u >> 31) | 0x80000000u);  // monotonic float->uint
}

__global__ __launch_bounds__(256) void sae_topk(
    const float* __restrict__ enc, float* __restrict__ out_w, int* __restrict__ out_i)
{
    const int row = blockIdx.x;
    const int tid = threadIdx.x;
    const float* e = enc + (size_t)row * NFEAT;

    __shared__ int hist[256];
    __shared__ unsigned s_prefix, s_mask;
    __shared__ int s_krem, c_gt, c_eq;
    __shared__ float sw[64];
    __shared__ int   si[64];

    if (tid == 0) { s_prefix = 0u; s_mask = 0u; s_krem = TOPK; c_gt = 0; c_eq = 0; }
    __syncthreads();

    for (int shift = 24; shift >= 0; shift -= 8) {
        hist[tid] = 0;
        __syncthreads();
        const unsigned pref = s_prefix, msk = s_mask;
        for (int i = tid; i < NFEAT; i += 256) {
            unsigned k = fkey(e[i]);
            if ((k & msk) == pref) atomicAdd(&hist[(k >> shift) & 255], 1);
        }
        __syncthreads();
        if (tid == 0) {
            int krem = s_krem, cum = 0, b;
            for (b = 255; b > 0; --b) {
                if (cum + hist[b] >= krem) break;
                cum += hist[b];
            }
            s_krem   = krem - cum;
            s_prefix = pref | ((unsigned)b << shift);
            s_mask   = msk | (0xFFu << shift);
        }
        __syncthreads();
    }

    const unsigned T = s_prefix;        // exact 64-th largest key
    const int krem   = s_krem;          // how many equal-to-T to take
    for (int i = tid; i < NFEAT; i += 256) {
        float v = e[i];
        unsigned k = fkey(v);
        if (k > T) {
            int p = atomicAdd(&c_gt, 1);
            sw[p] = v; si[p] = i;
        } else if (k == T) {
            int p = atomicAdd(&c_eq, 1);
            if (p < krem) { sw[63 - p] = v; si[63 - p] = i; }
        }
    }
    __syncthreads();

    // bitonic sort 64 elements: value desc, index asc (deterministic output order)
    for (int k = 2; k <= 64; k <<= 1) {
        for (int j = k >> 1; j > 0; j >>= 1) {
            if (tid < 64) {
                int ixj = tid ^ j;
                if (ixj > tid) {
                    float va = sw[tid], vb = sw[ixj];
                    int   ia = si[tid], ib = si[ixj];
                    unsigned ka = fkey(va), kb = fkey(vb);
                    bool aFirst = (ka > kb) || (ka == kb && ia < ib);
                    bool up = ((tid & k) == 0);
                    if (up ? !aFirst : aFirst) {
                        sw[tid] = vb; sw[ixj] = va;
                        si[tid] = ib; si[ixj] = ia;
                    }
                }
            }
            __syncthreads();
        }
    }
    if (tid < 64) {
        out_w[row * 64 + tid] = sw[tid];
        out_i[row * 64 + tid] = si[tid];
    }
}

// ---------------------------------------------------------------------------
// Kernel 3: top-32 of activated_in, lowest-index tie-break (matches jax top_k).
// ---------------------------------------------------------------------------
__global__ __launch_bounds__(256) void sae_dead_topk(
    const int* __restrict__ act, int* __restrict__ dead_i, int* __restrict__ dead_f)
{
    __shared__ unsigned taken[NFEAT / 32];   // 4KB bitmask
    __shared__ int bv[256], bi[256];
    const int tid = threadIdx.x;
    for (int i = tid; i < NFEAT / 32; i += 256) taken[i] = 0u;
    __syncthreads();

    for (int it = 0; it < AUXK; ++it) {
        int bestv = -2147483647 - 1, besti = -1;
        for (int i = tid; i < NFEAT; i += 256) {
            if (taken[i >> 5] & (1u << (i & 31))) continue;
            int v = act[i];
            if (v > bestv) { bestv = v; besti = i; }   // ascending scan => lowest idx on tie
        }
        bv[tid] = bestv; bi[tid] = besti;
        __syncthreads();
        for (int st = 128; st > 0; st >>= 1) {
            if (tid < st) {
                int v2 = bv[tid + st], i2 = bi[tid + st];
                if (v2 > bv[tid] || (v2 == bv[tid] && (unsigned)i2 < (unsigned)bi[tid])) {
                    bv[tid] = v2; bi[tid] = i2;
                }
            }
            __syncthreads();
        }
        if (tid == 0) {
            int w = bi[0];
            taken[w >> 5] |= (1u << (w & 31));
            dead_i[it] = w;
            dead_f[it] = (bv[0] > DEAD_AFTER) ? 1 : 0;
        }
        __syncthreads();
    }
}

// ---------------------------------------------------------------------------
// Kernel 4: sparse decode (top-64 + aux-32) and per-row loss.
// ---------------------------------------------------------------------------
__global__ __launch_bounds__(256) void sae_loss(
    const float* __restrict__ x, const float* __restrict__ W_dec,
    const float* __restrict__ b_post, const float* __restrict__ avg_norm,
    const float* __restrict__ enc, const float* __restrict__ tk_w,
    const int* __restrict__ tk_i, const int* __restrict__ dead_i,
    const int* __restrict__ dead_f, const int* __restrict__ n_steps,
    float* __restrict__ out)
{
    const int row = blockIdx.x;
    const int tid = threadIdx.x;
    __shared__ float w[64];  __shared__ int idx[64];
    __shared__ float dw[32]; __shared__ int didx[32];
    __shared__ float red[256], red2[256];

    if (tid < 64) {
        w[tid]   = tk_w[row * 64 + tid];
        idx[tid] = tk_i[row * 64 + tid];
    } else if (tid < 96) {
        int j  = tid - 64;
        int di = dead_i[j];
        didx[j] = di;
        dw[j]   = dead_f[j] ? enc[(size_t)row * NFEAT + di] : 0.f;
    }
    __syncthreads();

    const float s = sqrtf((float)DMODEL) / avg_norm[0];
    float ar = 0.f, aa = 0.f;
    for (int d = tid; d < DMODEL; d += 256) {
        float dec = 0.f, aux = 0.f;
#pragma unroll 16
        for (int j = 0; j < 64; ++j)
            dec = fmaf(w[j], W_dec[(size_t)idx[j] * DMODEL + d], dec);
#pragma unroll 16
        for (int j = 0; j < 32; ++j)
            aux = fmaf(dw[j], W_dec[(size_t)didx[j] * DMODEL + d], aux);
        float xn = x[(size_t)row * DMODEL + d] * s;
        float bp = b_post[d];
        float dr = xn - (dec + bp); ar = fmaf(dr, dr, ar);
        float da = xn - (aux + bp); aa = fmaf(da, da, aa);
    }
    red[tid] = ar; red2[tid] = aa;
    __syncthreads();
    for (int st = 128; st > 0; st >>= 1) {
        if (tid < st) { red[tid] += red[tid + st]; red2[tid] += red2[tid + st]; }
        __syncthreads();
    }
    if (tid == 0) {
        float l = red[0] / (float)DMODEL;
        if (n_steps[0] >= AUXK_AFTER)
            l += (1.0f / 32.0f) * (red2[0] / (float)DMODEL);
        out[row] = l;
    }
}

// ---------------------------------------------------------------------------
extern "C" void kernel_launch(void* const* d_in, const int* in_sizes, int n_in,
                              void* d_out, int out_size, void* d_ws, size_t ws_size,
                              hipStream_t stream)
{
    const float* x        = (const float*)d_in[0];
    const float* W_enc    = (const float*)d_in[1];
    const float* W_dec    = (const float*)d_in[2];
    const float* b_pre    = (const float*)d_in[3];
    const float* b_post   = (const float*)d_in[4];
    const float* avg_norm = (const float*)d_in[5];
    const int*   act_in   = (const int*)d_in[6];
    const int*   n_steps  = (const int*)d_in[7];
    float* out = (float*)d_out;

    // Workspace layout
    float* enc  = (float*)d_ws;                              // 2048*32768 f32 (256 MB)
    float* tk_w = enc + (size_t)BATCH * NFEAT;               // 2048*64 f32
    int*   tk_i = (int*)(tk_w + (size_t)BATCH * TOPK);       // 2048*64 i32
    int*   de_i = tk_i + (size_t)BATCH * TOPK;               // 32 i32
    int*   de_f = de_i + AUXK;                               // 32 i32
    float* cbuf = (float*)(de_f + AUXK);                     // 2048 f32

    sae_prep<<<DMODEL / 256, 256, 0, stream>>>(b_pre, b_post, cbuf);
    dim3 gGemm(NFEAT / 128, BATCH / 256);                    // (256, 8)
    sae_enc_gemm<<<gGemm, 256, 0, stream>>>(x, W_enc, cbuf, avg_norm, enc);
    sae_topk<<<BATCH, 256, 0, stream>>>(enc, tk_w, tk_i);
    sae_dead_topk<<<1, 256, 0, stream>>>(act_in, de_i, de_f);
    sae_loss<<<BATCH, 256, 0, stream>>>(x, W_dec, b_post, avg_norm, enc,
                                        tk_w, tk_i, de_i, de_f, n_steps, out);
}